// dDGM_60533269070284
// MI455X (gfx1250) — compile-verified
//
#include <hip/hip_runtime.h>
#include <hip/hip_bf16.h>
#include <float.h>

typedef __attribute__((ext_vector_type(2))) float v2f;
typedef __attribute__((ext_vector_type(8))) float v8f;
typedef __attribute__((ext_vector_type(4))) unsigned int u32x4;
typedef __attribute__((ext_vector_type(8))) int i32x8;
typedef __attribute__((ext_vector_type(4))) int i32x4;

#define N_PTS   8192
#define D_IN    128
#define D_EMB   64
#define KNN     16
#define CHUNK   128
#define CAND_MAX 128
#define LDPAD   68   // 64 + 4 pad: matches TDM pad_interval=64dw / pad_amount=4dw

// ---------------------------------------------------------------------------
// Kernel 1: X_tilde = X @ W + b   (fp32 WMMA 16x16x4)
// ---------------------------------------------------------------------------
__global__ __launch_bounds__(128) void embed_kernel(
    const float* __restrict__ X, const float* __restrict__ W,
    const float* __restrict__ bias, float* __restrict__ Xt)
{
    const int lane    = threadIdx.x & 31;
    const int wave    = threadIdx.x >> 5;        // 0..3 -> column tile of 16
    const int rowBase = blockIdx.x * 16;
    const int colBase = wave * 16;
    const int m  = lane & 15;
    const int kb = (lane >> 4) * 2;
    const int n  = lane & 15;

    v8f acc = {};
    #pragma unroll
    for (int k = 0; k < D_IN; k += 4) {
        v2f a, bf;
        a.x  = X[(rowBase + m) * D_IN + k + kb];
        a.y  = X[(rowBase + m) * D_IN + k + kb + 1];
        bf.x = W[(k + kb)     * D_EMB + colBase + n];
        bf.y = W[(k + kb + 1) * D_EMB + colBase + n];
        acc = __builtin_amdgcn_wmma_f32_16x16x4_f32(
            false, a, false, bf, (short)0, acc, false, false);
    }

    const int outCol = colBase + (lane & 15);
    const int rOff   = (lane >> 4) * 8;
    const float bv   = bias[outCol];
    #pragma unroll
    for (int i = 0; i < 8; ++i)
        Xt[(rowBase + rOff + i) * D_EMB + outCol] = acc[i] + bv;
}

// ---------------------------------------------------------------------------
// Kernel 2: sq[r] = sum_j X_tilde[r][j]^2
// ---------------------------------------------------------------------------
__global__ __launch_bounds__(256) void rowsq_kernel(
    const float* __restrict__ Xt, float* __restrict__ sq)
{
    int r = blockIdx.x * blockDim.x + threadIdx.x;
    if (r < N_PTS) {
        float s = 0.f;
        #pragma unroll
        for (int j = 0; j < D_EMB; ++j) { float v = Xt[r * D_EMB + j]; s += v * v; }
        sq[r] = s;
    }
}

// ---------------------------------------------------------------------------
// TDM: async-load one 128x64 f32 panel of X_tilde into LDS with a 4-DWORD pad
// every 64 DWORDs (-> row pitch LDPAD=68 floats). D# per CDNA5 ISA §8.3/8.4.
// Toolchain uses the 6-arg builtin: (u32x4, i32x8, i32x4, i32x4, i32x8, cpol).
// ---------------------------------------------------------------------------
#if defined(__has_builtin)
#if __has_builtin(__builtin_amdgcn_tensor_load_to_lds) && __has_builtin(__builtin_amdgcn_s_wait_tensorcnt)
#define USE_TDM 1
#endif
#endif

#ifdef USE_TDM
__device__ __forceinline__ void tdm_load_panel(const float* gsrc, unsigned int lds_off)
{
    const unsigned long long ga = (unsigned long long)(uintptr_t)gsrc;
    u32x4 g0;
    g0.x = 1u;                                   // count=1 (valid), load, user D#
    g0.y = lds_off;                              // lds_addr (bytes)
    g0.z = (unsigned int)ga;                     // global_addr[31:0]
    g0.w = (unsigned int)((ga >> 32) & 0x1FFFFFFu) | (2u << 30); // addr[56:32], type=2
    i32x8 g1;
    g1[0] = (int)((2u << 16)      // data_size = 4B
                | (1u << 20)      // pad_enable
                | (5u << 22)      // pad_interval: 64 DWORDs
                | (3u << 25));    // pad_amount : 4 DWORDs
    g1[1] = (int)(64u << 16);     // tensor_dim0 = 64 (lo16)
    g1[2] = (int)(8192u << 16);   // tensor_dim0 hi=0 | tensor_dim1 lo = 8192
    g1[3] = (int)(64u << 16);     // tensor_dim1 hi=0 | tile_dim0 = 64
    g1[4] = 128;                  // tile_dim1 = 128, tile_dim2 = 0
    g1[5] = 64;                   // tensor_dim0_stride = 64 elements (lo32)
    g1[6] = 0;                    // stride hi | dim1_stride lo (unused for 2D)
    g1[7] = 0;
    const i32x4 z4 = {0, 0, 0, 0};               // groups 2/3 unused (2D tensor)
    const i32x8 z8 = {0, 0, 0, 0, 0, 0, 0, 0};
    __builtin_amdgcn_tensor_load_to_lds(g0, g1, z4, z4, z8, 0);
}
#endif

// ---------------------------------------------------------------------------
// Kernel 3: fused Gram WMMA -> distance -> gumbel perturb -> streaming top-16,
// TDM double-buffered column panels. D never touches HBM.
// ---------------------------------------------------------------------------
__global__ __launch_bounds__(256) void knn_kernel(
    const float* __restrict__ Xt, const float* __restrict__ sq,
    const float* __restrict__ tptr, const float* __restrict__ q,
    int* __restrict__ edge, float* __restrict__ logprob)
{
    __shared__ float rowA[16][LDPAD];
    __shared__ float rowSq[16];
    __shared__ float colB[2][CHUNK][LDPAD];     // double-buffered panels
    __shared__ float colSq[CHUNK];
    __shared__ float topVal[16][KNN];
    __shared__ int   topIdx[16][KNN];
    __shared__ float topLog[16][KNN];
    __shared__ float cutoff[16];
    __shared__ float candVal[16][CAND_MAX];
    __shared__ int   candIdx[16][CAND_MAX];
    __shared__ float candLog[16][CAND_MAX];
    __shared__ int   candCnt[16];

    const int tid     = threadIdx.x;
    const int lane    = tid & 31;
    const int wave    = tid >> 5;
    const int rowBase = blockIdx.x * 16;

    float tv = tptr[0];
    tv = fminf(fmaxf(tv, -5.0f), 5.0f);
    const float expT = expf(tv);

    for (int idx = tid; idx < 16 * D_EMB; idx += 256)
        rowA[idx >> 6][idx & 63] = Xt[(rowBase + (idx >> 6)) * D_EMB + (idx & 63)];
    if (tid < 16) {
        rowSq[tid]   = sq[rowBase + tid];
        cutoff[tid]  = FLT_MAX;
        candCnt[tid] = 0;
    }
    {
        int r = tid >> 4, s = tid & 15;
        topVal[r][s] = FLT_MAX; topIdx[r][s] = 0; topLog[r][s] = 0.f;
    }

#ifdef USE_TDM
    if (wave == 0)   // TDM ignores EXEC; one wave issues for the whole block
        tdm_load_panel(Xt, (unsigned int)(uintptr_t)&colB[0][0][0]);
#endif
    __syncthreads();

    const int m  = lane & 15;
    const int kb = (lane >> 4) * 2;
    const int nn = lane & 15;
    const int n0 = wave * 16;
    int cur = 0;

    for (int c = 0; c < N_PTS / CHUNK; ++c) {
        const int colBase = c * CHUNK;

        if (tid < CHUNK) colSq[tid] = sq[colBase + tid];

#ifdef USE_TDM
        if (wave == 0) __builtin_amdgcn_s_wait_tensorcnt(0);  // panel `cur` landed
        __syncthreads();                                      // publish to all waves
        if (c + 1 < N_PTS / CHUNK && wave == 0)               // prefetch next panel
            tdm_load_panel(Xt + (size_t)(colBase + CHUNK) * D_EMB,
                           (unsigned int)(uintptr_t)&colB[cur ^ 1][0][0]);
#else
        {
            const float4* src = (const float4*)(Xt + (size_t)colBase * D_EMB);
            for (int idx = tid; idx < CHUNK * (D_EMB / 4); idx += 256) {
                int row = idx >> 4;
                ((float4*)&colB[cur][row][0])[idx & 15] = src[idx];
            }
        }
        __syncthreads();
#endif

        // ---- 16x16 Gram tile per wave: G = rows · colsᵀ, K = 64 ----
        v8f acc = {};
        #pragma unroll
        for (int k = 0; k < D_EMB; k += 4) {
            v2f a, bf;
            a.x  = rowA[m][k + kb];
            a.y  = rowA[m][k + kb + 1];
            bf.x = colB[cur][n0 + nn][k + kb];
            bf.y = colB[cur][n0 + nn][k + kb + 1];
            acc = __builtin_amdgcn_wmma_f32_16x16x4_f32(
                false, a, false, bf, (short)0, acc, false, false);
        }

        // ---- distances -> perturbed logits straight out of the C layout ----
        const int rOff = (lane >> 4) * 8;
        const int gcol = colBase + n0 + nn;
        const float sc = colSq[n0 + nn];
        #pragma unroll
        for (int i = 0; i < 8; ++i) {
            const int mloc = rOff + i;
            float d = rowSq[mloc] + sc - 2.0f * acc[i];
            d = fmaxf(d, 0.0f);
            const float logit = d * expT;
            const float qv    = q[(size_t)(rowBase + mloc) * N_PTS + gcol];
            const float pert  = logit + logf(-logf(qv));   // logits - gumbel
            if (pert < cutoff[mloc]) {
                int p = atomicAdd(&candCnt[mloc], 1);
                if (p < CAND_MAX) {
                    candVal[mloc][p] = pert;
                    candIdx[mloc][p] = gcol;
                    candLog[mloc][p] = logit;
                }
            }
        }
        __syncthreads();

        // ---- merge candidates into running top-16 (one thread per row) ----
        if (tid < 16) {
            const int r   = tid;
            const int cnt = min(candCnt[r], CAND_MAX);
            for (int j = 0; j < cnt; ++j) {
                const float v = candVal[r][j];
                int am = 0; float mv = topVal[r][0];
                #pragma unroll
                for (int s = 1; s < KNN; ++s)
                    if (topVal[r][s] > mv) { mv = topVal[r][s]; am = s; }
                if (v < mv) {
                    topVal[r][am] = v;
                    topIdx[r][am] = candIdx[r][j];
                    topLog[r][am] = candLog[r][j];
                }
            }
            float mv = topVal[r][0];
            #pragma unroll
            for (int s = 1; s < KNN; ++s) mv = fmaxf(mv, topVal[r][s]);
            cutoff[r]  = mv;
            candCnt[r] = 0;
        }
        __syncthreads();
        cur ^= 1;
    }

    // ---- sort ascending by perturbed value; emit edges + logprobs ----
    if (tid < 16) {
        const int r    = tid;
        const int grow = rowBase + r;
        for (int i = 1; i < KNN; ++i) {
            float kv = topVal[r][i]; int ki = topIdx[r][i]; float kl = topLog[r][i];
            int j = i - 1;
            while (j >= 0 && topVal[r][j] > kv) {
                topVal[r][j + 1] = topVal[r][j];
                topIdx[r][j + 1] = topIdx[r][j];
                topLog[r][j + 1] = topLog[r][j];
                --j;
            }
            topVal[r][j + 1] = kv; topIdx[r][j + 1] = ki; topLog[r][j + 1] = kl;
        }
        for (int k = 0; k < KNN; ++k) {
            const int e = grow * KNN + k;
            edge[e]               = topIdx[r][k];   // edge_index[0]
            edge[N_PTS * KNN + e] = grow;           // edge_index[1]
            logprob[e]            = -topLog[r][k];
        }
    }
}

// ---------------------------------------------------------------------------
extern "C" void kernel_launch(void* const* d_in, const int* in_sizes, int n_in,
                              void* d_out, int out_size, void* d_ws, size_t ws_size,
                              hipStream_t stream) {
    const float* X    = (const float*)d_in[0];   // [1,8192,128]
    const float* W    = (const float*)d_in[1];   // [128,64]
    const float* bias = (const float*)d_in[2];   // [64]
    const float* t    = (const float*)d_in[3];   // scalar
    const float* q    = (const float*)d_in[4];   // [1,8192,8192]

    float* Xt      = (float*)d_out;                               // 8192*64 f32
    int*   edge    = (int*)((float*)d_out + N_PTS * D_EMB);       // 2*8192*16 i32
    float* logprob = (float*)d_out + N_PTS * D_EMB + 2 * N_PTS * KNN;
    float* sq      = (float*)d_ws;                                // 8192 f32

    embed_kernel<<<N_PTS / 16, 128, 0, stream>>>(X, W, bias, Xt);
    rowsq_kernel<<<N_PTS / 256, 256, 0, stream>>>(Xt, sq);
    knn_kernel<<<N_PTS / 16, 256, 0, stream>>>(Xt, sq, t, q, edge, logprob);
}